// MambaBlockSSM_18562848653363
// MI455X (gfx1250) — compile-verified
//
#include <hip/hip_runtime.h>
#include <hip/hip_bf16.h>
#include <math.h>
#include <stdint.h>

// ---------------------------------------------------------------------------
// Mamba block for MI455X (gfx1250, wave32, WMMA f32 16x16x4, async->LDS)
//   B=2, L=2048, D_MODEL=1024, D_INNER=2048, D_STATE=16, DT_RANK=64, D_CONV=4
// ---------------------------------------------------------------------------

#define B_SZ      2
#define L_SZ      2048
#define D_MODEL   1024
#define D_INNER   2048
#define D_STATE   16
#define DT_RANK   64
#define D_CONV    4
#define NTOK      (B_SZ * L_SZ)            // 4096 tokens
#define XPROJ_N   (DT_RANK + 2 * D_STATE)  // 96

// 1 = use GLOBAL_LOAD_ASYNC_TO_LDS_B128 (CDNA5 async DMA path, ASYNCcnt)
// 0 = register-staged global->LDS fallback
#ifndef USE_ASYNC_LDS
#define USE_ASYNC_LDS 1
#endif

typedef __attribute__((ext_vector_type(2))) float v2f;
typedef __attribute__((ext_vector_type(8))) float v8f;

__device__ __forceinline__ float silu_f(float x) {
    return x / (1.0f + __expf(-x));
}
__device__ __forceinline__ float softplus_f(float x) {
    return (x > 20.0f) ? x : log1pf(__expf(x));
}

// ---------------------------------------------------------------------------
// RMSNorm: one block (256 threads) per token; 1024 floats per token.
// ---------------------------------------------------------------------------
__global__ void rmsnorm_kernel(const float* __restrict__ x,
                               const float* __restrict__ w,
                               float* __restrict__ h) {
    const int token = blockIdx.x;
    const int tid   = threadIdx.x;
    const float4* xr = (const float4*)(x + (size_t)token * D_MODEL);
    float4 v = xr[tid];
    float ss = v.x * v.x + v.y * v.y + v.z * v.z + v.w * v.w;
    ss += __shfl_xor(ss, 1);
    ss += __shfl_xor(ss, 2);
    ss += __shfl_xor(ss, 4);
    ss += __shfl_xor(ss, 8);
    ss += __shfl_xor(ss, 16);
    __shared__ float part[8];
    if ((tid & 31) == 0) part[tid >> 5] = ss;
    __syncthreads();
    float tot = 0.0f;
#pragma unroll
    for (int i = 0; i < 8; ++i) tot += part[i];
    const float r = rsqrtf(tot * (1.0f / D_MODEL) + 1e-5f);
    const float4* wr = (const float4*)w;
    float4 wv = wr[tid];
    float4 o;
    o.x = v.x * r * wv.x; o.y = v.y * r * wv.y;
    o.z = v.z * r * wv.z; o.w = v.w * r * wv.w;
    ((float4*)(h + (size_t)token * D_MODEL))[tid] = o;
}

// ---------------------------------------------------------------------------
// Depthwise causal conv1d + bias + SiLU.
// ---------------------------------------------------------------------------
__global__ void conv_silu_kernel(const float* __restrict__ xz,
                                 const float* __restrict__ cw,
                                 const float* __restrict__ cb,
                                 float* __restrict__ out) {
    const int idx = blockIdx.x * blockDim.x + threadIdx.x;
    if (idx >= NTOK * D_INNER) return;
    const int d     = idx & (D_INNER - 1);
    const int token = idx >> 11;
    const int b     = token >> 11;
    const int l     = token & (L_SZ - 1);
    float acc = cb[d];
#pragma unroll
    for (int j = 0; j < D_CONV; ++j) {
        const int ll = l - (D_CONV - 1) + j;
        if (ll >= 0) {
            acc += cw[d * D_CONV + j] *
                   xz[(size_t)(b * L_SZ + ll) * (2 * D_INNER) + d];
        }
    }
    out[(size_t)token * D_INNER + d] = silu_f(acc);
}

// ---------------------------------------------------------------------------
// WMMA f32 GEMM:  C[m,n] = sum_k A[m,k] * W[n,k]
//   MODE 0: plain store
//   MODE 1: C = softplus(acc + bias[n])       (dt_proj)
//   MODE 2: C = acc + res[m,n]                (out_proj; gating fused in scan)
//
// Block = 256 threads = 8 wave32; tile 64x128; BK=16; double-buffered LDS.
// Stage k+1 streams global->LDS via GLOBAL_LOAD_ASYNC_TO_LDS_B128 while the
// 16 WMMAs of stage k run from LDS; drained with s_wait_asynccnt + barrier.
// M % 64 == 0 and K % 16 == 0 for all four GEMMs; N guarded (x_proj N=96).
// ---------------------------------------------------------------------------
template <int MODE>
__global__ void gemm_wmma_kernel(const float* __restrict__ A,
                                 const float* __restrict__ W,
                                 const float* __restrict__ extra,
                                 float* __restrict__ C,
                                 int M, int N, int K,
                                 int lda, int ldw, int ldc) {
    constexpr int BK   = 16;
    constexpr int ASTR = 20;   // LDS row stride: 80B -> b128 writes 16B aligned
    __shared__ float As[2][64 * ASTR];
    __shared__ float Bs[2][128 * ASTR];

    const int tid  = threadIdx.x;
    const int wave = tid >> 5;
    const int lane = tid & 31;
    const int lo   = lane & 15;
    const int hi   = lane >> 4;

    const int m0   = blockIdx.x * 64;
    const int nblk = blockIdx.y * 128;
    const int ncol = nblk + wave * 16 + lo;

    // cooperative fill mapping: one 16B chunk of A, two of W per thread
    const int fr = tid >> 2;            // 0..63
    const int fc = (tid & 3) << 2;      // 0,4,8,12
    const int wrow0 = nblk + fr;
    const int wrow1 = nblk + 64 + fr;
    const float* aSrc  = A + (size_t)(m0 + fr) * lda + fc;
    const float* wSrc0 = W + (size_t)((wrow0 < N) ? wrow0 : (N - 1)) * ldw + fc;
    const float* wSrc1 = W + (size_t)((wrow1 < N) ? wrow1 : (N - 1)) * ldw + fc;

#if !USE_ASYNC_LDS
    float4 aReg, wReg0, wReg1;
#endif

    auto issue = [&](int k0, int buf) {
#if USE_ASYNC_LDS
        asm volatile("global_load_async_to_lds_b128 %0, %1, off"
                     :: "v"((uint32_t)(uintptr_t)&As[buf][fr * ASTR + fc]),
                        "v"(aSrc + k0) : "memory");
        asm volatile("global_load_async_to_lds_b128 %0, %1, off"
                     :: "v"((uint32_t)(uintptr_t)&Bs[buf][fr * ASTR + fc]),
                        "v"(wSrc0 + k0) : "memory");
        asm volatile("global_load_async_to_lds_b128 %0, %1, off"
                     :: "v"((uint32_t)(uintptr_t)&Bs[buf][(64 + fr) * ASTR + fc]),
                        "v"(wSrc1 + k0) : "memory");
#else
        aReg  = *(const float4*)(aSrc + k0);
        wReg0 = *(const float4*)(wSrc0 + k0);
        wReg1 = *(const float4*)(wSrc1 + k0);
#endif
    };
    auto finalize = [&](int buf) {
#if USE_ASYNC_LDS
        asm volatile("s_wait_asynccnt 0x0" ::: "memory");
#else
        *(float4*)&As[buf][fr * ASTR + fc]        = aReg;
        *(float4*)&Bs[buf][fr * ASTR + fc]        = wReg0;
        *(float4*)&Bs[buf][(64 + fr) * ASTR + fc] = wReg1;
#endif
    };

    v8f acc0 = {}, acc1 = {}, acc2 = {}, acc3 = {};
    const int nst = K / BK;

    issue(0, 0);
    finalize(0);
    __syncthreads();

    int cur = 0;
    for (int ks = 0; ks < nst; ++ks) {
        const int  nxt  = cur ^ 1;
        const bool more = (ks + 1) < nst;
        if (more) issue((ks + 1) * BK, nxt);

        // fragments for the whole stage first, then 16 back-to-back WMMAs
        v2f bf[4], af[4][4];
        const float* asb = &As[cur][0];
        const float* bsb = &Bs[cur][0];
#pragma unroll
        for (int q = 0; q < 4; ++q) {
            const int kk = q * 4 + (hi << 1);
            bf[q] = *(const v2f*)&bsb[(wave * 16 + lo) * ASTR + kk];
#pragma unroll
            for (int t = 0; t < 4; ++t)
                af[q][t] = *(const v2f*)&asb[(t * 16 + lo) * ASTR + kk];
        }
#pragma unroll
        for (int q = 0; q < 4; ++q) {
            acc0 = __builtin_amdgcn_wmma_f32_16x16x4_f32(false, af[q][0], false, bf[q],
                                                         (short)0, acc0, false, false);
            acc1 = __builtin_amdgcn_wmma_f32_16x16x4_f32(false, af[q][1], false, bf[q],
                                                         (short)0, acc1, false, false);
            acc2 = __builtin_amdgcn_wmma_f32_16x16x4_f32(false, af[q][2], false, bf[q],
                                                         (short)0, acc2, false, false);
            acc3 = __builtin_amdgcn_wmma_f32_16x16x4_f32(false, af[q][3], false, bf[q],
                                                         (short)0, acc3, false, false);
        }

        if (more) finalize(nxt);
        __syncthreads();
        cur = nxt;
    }

    if (ncol >= N) return;
    v8f accs[4] = {acc0, acc1, acc2, acc3};
#pragma unroll
    for (int t = 0; t < 4; ++t) {
        const int rbase = m0 + t * 16 + hi * 8;   // C/D layout: lanes16-31 -> M+8
#pragma unroll
        for (int i = 0; i < 8; ++i) {
            float v = accs[t][i];
            const int row = rbase + i;
            if (MODE == 1) v = softplus_f(v + extra[ncol]);
            if (MODE == 2) v += extra[(size_t)row * ldc + ncol];
            C[(size_t)row * ldc + ncol] = v;
        }
    }
}

// ---------------------------------------------------------------------------
// Selective scan. 16 lanes per (b,d) channel, one lane per state index n.
// Fused epilogue: y = (state.C + x*D_skip) * silu(gate).
// ---------------------------------------------------------------------------
__global__ void scan_kernel(const float* __restrict__ delta,  // [NTOK, D_INNER]
                            const float* __restrict__ xsc,    // [NTOK, D_INNER]
                            const float* __restrict__ xp,     // [NTOK, 96]
                            const float* __restrict__ xz,     // [NTOK, 2*D_INNER]
                            const float* __restrict__ A_log,  // [D_INNER, 16]
                            const float* __restrict__ Dsk,    // [D_INNER]
                            float* __restrict__ y) {          // [NTOK, D_INNER]
    const int ch = blockIdx.x * 16 + (threadIdx.x >> 4);
    const int n  = threadIdx.x & 15;
    const int b  = ch >> 11;
    const int d  = ch & (D_INNER - 1);

    const float An = -__expf(A_log[d * D_STATE + n]);
    const float Dd = Dsk[d];

    const size_t tok0 = (size_t)b * L_SZ;
    const float* dp = delta + tok0 * D_INNER + d;
    const float* xq = xsc   + tok0 * D_INNER + d;
    const float* bp = xp + tok0 * XPROJ_N + DT_RANK + n;
    const float* cp = xp + tok0 * XPROJ_N + DT_RANK + D_STATE + n;
    const float* gp = xz + tok0 * (2 * D_INNER) + D_INNER + d;  // gate
    float*       yp = y + tok0 * D_INNER + d;

    float s = 0.0f;
    for (int l = 0; l < L_SZ; ++l) {
        const float dl = *dp;
        const float xv = *xq;
        const float Bn = *bp;
        const float Cn = *cp;
        const float dA = __expf(dl * An);
        s = fmaf(dA, s, dl * xv * Bn);
        float t = s * Cn;
        t += __shfl_xor(t, 1);
        t += __shfl_xor(t, 2);
        t += __shfl_xor(t, 4);
        t += __shfl_xor(t, 8);
        if (n == 0) {
            const float g = *gp;
            *yp = (t + xv * Dd) * silu_f(g);
        }
        dp += D_INNER; xq += D_INNER; yp += D_INNER;
        bp += XPROJ_N; cp += XPROJ_N; gp += 2 * D_INNER;
    }
}

// ---------------------------------------------------------------------------
extern "C" void kernel_launch(void* const* d_in, const int* in_sizes, int n_in,
                              void* d_out, int out_size, void* d_ws, size_t ws_size,
                              hipStream_t stream) {
    const float* x        = (const float*)d_in[0];   // [2,2048,1024]
    const float* norm_w   = (const float*)d_in[1];   // [1024]
    const float* in_w     = (const float*)d_in[2];   // [4096,1024]
    const float* conv_w   = (const float*)d_in[3];   // [2048,1,4]
    const float* conv_b   = (const float*)d_in[4];   // [2048]
    const float* xproj_w  = (const float*)d_in[5];   // [96,2048]
    const float* dt_w     = (const float*)d_in[6];   // [2048,64]
    const float* dt_b     = (const float*)d_in[7];   // [2048]
    const float* A_log    = (const float*)d_in[8];   // [2048,16]
    const float* D_skip   = (const float*)d_in[9];   // [2048]
    const float* out_w    = (const float*)d_in[10];  // [1024,2048]
    float* out = (float*)d_out;                      // [2,2048,1024]

    // workspace layout (floats)
    float* ws    = (float*)d_ws;
    float* h     = ws;                                   // 4096*1024
    float* xz    = h     + (size_t)NTOK * D_MODEL;       // 4096*4096
    float* xsc   = xz    + (size_t)NTOK * 2 * D_INNER;   // 4096*2048
    float* xp    = xsc   + (size_t)NTOK * D_INNER;       // 4096*96
    float* delta = xp    + (size_t)NTOK * XPROJ_N;       // 4096*2048
    float* y     = delta + (size_t)NTOK * D_INNER;       // 4096*2048

    // 1) RMSNorm
    rmsnorm_kernel<<<NTOK, 256, 0, stream>>>(x, norm_w, h);

    // 2) in_proj: xz[4096,4096] = h @ in_w^T
    gemm_wmma_kernel<0><<<dim3(NTOK / 64, 32), 256, 0, stream>>>(
        h, in_w, nullptr, xz,
        NTOK, 2 * D_INNER, D_MODEL, D_MODEL, D_MODEL, 2 * D_INNER);

    // 3) depthwise causal conv + SiLU -> xsc[4096,2048]
    conv_silu_kernel<<<(NTOK * D_INNER) / 256, 256, 0, stream>>>(
        xz, conv_w, conv_b, xsc);

    // 4) x_proj: xp[4096,96] = xsc @ xproj_w^T
    gemm_wmma_kernel<0><<<dim3(NTOK / 64, 1), 256, 0, stream>>>(
        xsc, xproj_w, nullptr, xp,
        NTOK, XPROJ_N, D_INNER, D_INNER, D_INNER, XPROJ_N);

    // 5) dt_proj + bias + softplus: delta = sp(xp[:, :64] @ dt_w^T + b)
    gemm_wmma_kernel<1><<<dim3(NTOK / 64, D_INNER / 128), 256, 0, stream>>>(
        xp, dt_w, dt_b, delta,
        NTOK, D_INNER, DT_RANK, XPROJ_N, DT_RANK, D_INNER);

    // 6) selective scan (+ D_skip, + silu(gate)) -> y[4096,2048]
    scan_kernel<<<(B_SZ * D_INNER) / 16, 256, 0, stream>>>(
        delta, xsc, xp, xz, A_log, D_skip, y);

    // 7) out_proj + residual: out = x + y @ out_w^T
    gemm_wmma_kernel<2><<<dim3(NTOK / 64, D_MODEL / 128), 256, 0, stream>>>(
        y, out_w, x, out,
        NTOK, D_MODEL, D_INNER, D_INNER, D_INNER, D_MODEL);
}